// LogArapEnergy_46059229282949
// MI455X (gfx1250) — compile-verified
//
#include <hip/hip_runtime.h>
#include <hip/hip_bf16.h>
#include <math.h>

// ---------------------------------------------------------------------------
// Problem constants (from reference setup_inputs)
// ---------------------------------------------------------------------------
#define NVERT 20000
#define KNBR  16
#define DCODE 256
#define H1DIM 512
#define H2DIM 1024
#define M3    60000   // NVERT * 3

// workspace layout (floats)
#define OFF_H1    0
#define OFF_H2    (OFF_H1 + H1DIM)            // 512
#define OFF_RECON (OFF_H2 + H2DIM)            // 1536
#define OFF_GV    (OFF_RECON + M3)            // 61536
#define OFF_GH2   (OFF_GV + M3)               // 121536
#define OFF_EN    (OFF_GH2 + H2DIM)           // 122560
#define OFF_GH1   (OFF_EN + 1)                // 122561

typedef __attribute__((ext_vector_type(2))) float v2f;
typedef __attribute__((ext_vector_type(8))) float v8f;

// ---------------------------------------------------------------------------
// zero scratch accumulators
// ---------------------------------------------------------------------------
__global__ void zero_kernel(float* __restrict__ p, int n) {
  int i = blockIdx.x * blockDim.x + threadIdx.x;
  if (i < n) p[i] = 0.0f;
}

// ---------------------------------------------------------------------------
// small matvec + relu:  y[o] = relu(b[o] + sum_i x[i] * W[i*out + o])
// ---------------------------------------------------------------------------
__global__ void mlp_mv_relu(const float* __restrict__ x, const float* __restrict__ W,
                            const float* __restrict__ b, float* __restrict__ y,
                            int in_dim, int out_dim) {
  int o = blockIdx.x * blockDim.x + threadIdx.x;
  if (o >= out_dim) return;
  float acc = b[o];
  for (int i = 0; i < in_dim; ++i)
    acc = fmaf(x[i], W[(size_t)i * out_dim + o], acc);
  y[o] = fmaxf(acc, 0.0f);
}

// ---------------------------------------------------------------------------
// forward big matvec via WMMA f32 16x16x4:
//   recon[j] = b3[j] + sum_k h2[k] * W3[k*M3 + j]
// One wave per 16-column tile. A holds h2 in row M=0, B holds 4x16 tile of W3.
// ---------------------------------------------------------------------------
__global__ void __launch_bounds__(256)
decode_wmma(const float* __restrict__ h2, const float* __restrict__ W3,
            const float* __restrict__ b3, float* __restrict__ recon) {
  int wave = (int)((blockIdx.x * blockDim.x + threadIdx.x) >> 5);
  int lane = threadIdx.x & 31;
  if (wave >= (M3 / 16)) return;                 // wave-uniform guard
  const int j0   = wave * 16;
  const int half = lane >> 4;                    // 0: K={0,1}, 1: K={2,3}
  const int l15  = lane & 15;

  v8f c = {};
  const float* bp = W3 + (size_t)(2 * half) * M3 + j0 + l15;
  for (int k = 0; k < H2DIM; k += 4) {
    // A-matrix 16x4: only row M=0 is live (lanes 0 and 16 carry h2)
    float ha = h2[k + 2 * half];
    float hb = h2[k + 2 * half + 1];
    v2f a;
    a.x = (l15 == 0) ? ha : 0.0f;
    a.y = (l15 == 0) ? hb : 0.0f;
    // B-matrix 4x16 tile of W3 (rows k..k+3, cols j0..j0+15)
    v2f b;
    b.x = bp[0];
    b.y = bp[M3];
    bp += (size_t)4 * M3;
    __builtin_prefetch(bp, 0, 0);                // stream-ahead hint (global_prefetch_b8)
    c = __builtin_amdgcn_wmma_f32_16x16x4_f32(false, a, false, b, (short)0, c,
                                              false, false);
  }
  // result row M=0 lives in C VGPR0, lanes 0..15 (N = lane)
  if (half == 0) recon[j0 + l15] = c[0] + b3[j0 + l15];
}

// ---------------------------------------------------------------------------
// backward big matvec via WMMA f32 16x16x4:
//   gh2[i] += sum_j W3[i*M3 + j] * gv[j]
// A = 16x4 tile of W3 (rows i0..i0+15, cols j..j+3, contiguous b64 per lane),
// B = gv in column N=0. Partials in C column 0 -> atomics into gh2.
// ---------------------------------------------------------------------------
#define BW_CHUNKS 30
#define BW_STEPS  500   // 500 * 4 cols = 2000 cols per chunk; 30*2000 = 60000

__global__ void __launch_bounds__(256)
bwd_w3_wmma(const float* __restrict__ W3, const float* __restrict__ gv,
            float* __restrict__ gh2) {
  int wave = (int)((blockIdx.x * blockDim.x + threadIdx.x) >> 5);
  int lane = threadIdx.x & 31;
  if (wave >= (H2DIM / 16) * BW_CHUNKS) return;  // wave-uniform guard
  const int itile = wave / BW_CHUNKS;            // 0..63
  const int chunk = wave % BW_CHUNKS;            // 0..29
  const int i0    = itile * 16;
  const int jbase = chunk * (BW_STEPS * 4);
  const int half  = lane >> 4;
  const int l15   = lane & 15;

  v8f c = {};
  const float* ap = W3 + (size_t)(i0 + l15) * M3 + jbase + 2 * half;
  for (int s = 0; s < BW_STEPS; ++s) {
    const int j = jbase + 4 * s;
    v2f a;                                       // contiguous pair -> b64 load
    a.x = ap[0];
    a.y = ap[1];
    ap += 4;
    __builtin_prefetch(ap + 60, 0, 0);
    float g0 = gv[j + 2 * half];
    float g1 = gv[j + 2 * half + 1];
    v2f b;
    b.x = (l15 == 0) ? g0 : 0.0f;                // B column N=0 = grad_v
    b.y = (l15 == 0) ? g1 : 0.0f;
    c = __builtin_amdgcn_wmma_f32_16x16x4_f32(false, a, false, b, (short)0, c,
                                              false, false);
  }
  // C column N=0: lane 0 holds M=0..7 in VGPR 0..7, lane 16 holds M=8..15
  if (l15 == 0) {
    const int base = i0 + half * 8;
#pragma unroll
    for (int r = 0; r < 8; ++r) atomicAdd(&gh2[base + r], c[r]);
  }
}

// ---------------------------------------------------------------------------
// ARAP per-vertex: covariance, 3x3 SVD -> rotation (stop-grad), energy + dE/drecon
// ---------------------------------------------------------------------------
#define JROT(p, q)                                                             \
  {                                                                            \
    float apq = Bm[p][q];                                                      \
    if (fabsf(apq) > 1e-12f) {                                                 \
      float app = Bm[p][p], aqq = Bm[q][q];                                    \
      float tau = (aqq - app) / (2.0f * apq);                                  \
      float t = ((tau >= 0.0f) ? 1.0f : -1.0f) /                               \
                (fabsf(tau) + sqrtf(1.0f + tau * tau));                        \
      float cth = 1.0f / sqrtf(1.0f + t * t);                                  \
      float sth = t * cth;                                                     \
      _Pragma("unroll") for (int r = 0; r < 3; ++r) {                          \
        float brp = Bm[r][p], brq = Bm[r][q];                                  \
        Bm[r][p] = cth * brp - sth * brq;                                      \
        Bm[r][q] = sth * brp + cth * brq;                                      \
      }                                                                        \
      _Pragma("unroll") for (int r = 0; r < 3; ++r) {                          \
        float bpr = Bm[p][r], bqr = Bm[q][r];                                  \
        Bm[p][r] = cth * bpr - sth * bqr;                                      \
        Bm[q][r] = sth * bpr + cth * bqr;                                      \
      }                                                                        \
      _Pragma("unroll") for (int r = 0; r < 3; ++r) {                          \
        float vrp = Vm[r][p], vrq = Vm[r][q];                                  \
        Vm[r][p] = cth * vrp - sth * vrq;                                      \
        Vm[r][q] = sth * vrp + cth * vrq;                                      \
      }                                                                        \
    }                                                                          \
  }

#define CSWAP(p, q)                                                            \
  if (e##p < e##q) {                                                           \
    float te = e##p; e##p = e##q; e##q = te;                                   \
    _Pragma("unroll") for (int r = 0; r < 3; ++r) {                            \
      float tv = Vm[r][p]; Vm[r][p] = Vm[r][q]; Vm[r][q] = tv;                 \
    }                                                                          \
  }

__global__ void __launch_bounds__(256)
arap_kernel(const float* __restrict__ xyz1, const float* __restrict__ recon,
            const float* __restrict__ wmat, const int* __restrict__ nbr,
            const int* __restrict__ nnbr, float* __restrict__ grad_v,
            float* __restrict__ energy) {
  int i = blockIdx.x * blockDim.x + threadIdx.x;
  if (i >= NVERT) return;
  const int nn = nnbr[i];
  float xi[3], ri[3];
#pragma unroll
  for (int c = 0; c < 3; ++c) { xi[c] = xyz1[3 * i + c]; ri[c] = recon[3 * i + c]; }

  // --- pass 1: weighted covariance S = sum_k w * e_rest e_def^T ---
  float S[3][3] = {{0, 0, 0}, {0, 0, 0}, {0, 0, 0}};
  for (int k = 0; k < KNBR; ++k) {
    float w = (k < nn) ? wmat[i * KNBR + k] : 0.0f;
    int j = nbr[i * KNBR + k];
    float er[3], ed[3];
#pragma unroll
    for (int c = 0; c < 3; ++c) {
      er[c] = xyz1[3 * j + c] - xi[c];
      ed[c] = recon[3 * j + c] - ri[c];
    }
#pragma unroll
    for (int a = 0; a < 3; ++a)
#pragma unroll
      for (int b = 0; b < 3; ++b) S[a][b] = fmaf(w * er[a], ed[b], S[a][b]);
  }

  // --- SVD of S via Jacobi on B = S^T S  (S = U Sigma V^T) ---
  float Bm[3][3], Vm[3][3] = {{1, 0, 0}, {0, 1, 0}, {0, 0, 1}};
#pragma unroll
  for (int a = 0; a < 3; ++a)
#pragma unroll
    for (int b = 0; b < 3; ++b)
      Bm[a][b] = S[0][a] * S[0][b] + S[1][a] * S[1][b] + S[2][a] * S[2][b];
#pragma unroll
  for (int sweep = 0; sweep < 4; ++sweep) { JROT(0, 1) JROT(0, 2) JROT(1, 2) }
  float e0 = Bm[0][0], e1 = Bm[1][1], e2 = Bm[2][2];
  CSWAP(0, 1) CSWAP(1, 2) CSWAP(0, 1)     // sort eigenpairs descending
  (void)e0; (void)e1; (void)e2;

  // U columns: u_i = S v_i / sigma_i  (with degeneracy guards)
  float u0[3], u1[3], u2[3];
#pragma unroll
  for (int a = 0; a < 3; ++a) {
    u0[a] = S[a][0] * Vm[0][0] + S[a][1] * Vm[1][0] + S[a][2] * Vm[2][0];
    u1[a] = S[a][0] * Vm[0][1] + S[a][1] * Vm[1][1] + S[a][2] * Vm[2][1];
  }
  float n0 = sqrtf(u0[0] * u0[0] + u0[1] * u0[1] + u0[2] * u0[2]);
  if (n0 > 1e-12f) { float inv = 1.0f / n0; u0[0] *= inv; u0[1] *= inv; u0[2] *= inv; }
  else { u0[0] = 1.0f; u0[1] = 0.0f; u0[2] = 0.0f; }
  float d01 = u0[0] * u1[0] + u0[1] * u1[1] + u0[2] * u1[2];
#pragma unroll
  for (int a = 0; a < 3; ++a) u1[a] -= d01 * u0[a];
  float n1 = sqrtf(u1[0] * u1[0] + u1[1] * u1[1] + u1[2] * u1[2]);
  if (n1 > 1e-12f) { float inv = 1.0f / n1; u1[0] *= inv; u1[1] *= inv; u1[2] *= inv; }
  else {  // any unit vector orthogonal to u0
    float t[3] = {0.0f, 1.0f, 0.0f};
    if (fabsf(u0[1]) > 0.9f) { t[1] = 0.0f; t[2] = 1.0f; }
    float dt = u0[0] * t[0] + u0[1] * t[1] + u0[2] * t[2];
#pragma unroll
    for (int a = 0; a < 3; ++a) u1[a] = t[a] - dt * u0[a];
    float nn1 = sqrtf(u1[0] * u1[0] + u1[1] * u1[1] + u1[2] * u1[2]) + 1e-20f;
#pragma unroll
    for (int a = 0; a < 3; ++a) u1[a] /= nn1;
  }
  u2[0] = u0[1] * u1[2] - u0[2] * u1[1];
  u2[1] = u0[2] * u1[0] - u0[0] * u1[2];
  u2[2] = u0[0] * u1[1] - u0[1] * u1[0];

  // R0 = V U^T ; reflection fix R = V diag(1,1,det(R0)) U^T
  float R0[3][3];
#pragma unroll
  for (int a = 0; a < 3; ++a)
#pragma unroll
    for (int c = 0; c < 3; ++c)
      R0[a][c] = Vm[a][0] * u0[c] + Vm[a][1] * u1[c] + Vm[a][2] * u2[c];
  float det = R0[0][0] * (R0[1][1] * R0[2][2] - R0[1][2] * R0[2][1]) -
              R0[0][1] * (R0[1][0] * R0[2][2] - R0[1][2] * R0[2][0]) +
              R0[0][2] * (R0[1][0] * R0[2][1] - R0[1][1] * R0[2][0]);
  float dsel = (det < 0.0f) ? -1.0f : 1.0f;
  float R[3][3];
#pragma unroll
  for (int a = 0; a < 3; ++a)
#pragma unroll
    for (int c = 0; c < 3; ++c) R[a][c] = R0[a][c] + (dsel - 1.0f) * Vm[a][2] * u2[c];

  // --- pass 2: energy + scatter gradient of E = (1/N) sum_i pv_i ---
  float acc[3] = {0, 0, 0};
  float pv = 0.0f;
  const float gscale = 2.0f / (float)NVERT;
  for (int k = 0; k < KNBR; ++k) {
    float w = (k < nn) ? wmat[i * KNBR + k] : 0.0f;
    int j = nbr[i * KNBR + k];
    float er[3], ed[3], res[3];
#pragma unroll
    for (int c = 0; c < 3; ++c) {
      er[c] = xyz1[3 * j + c] - xi[c];
      ed[c] = recon[3 * j + c] - ri[c];
    }
#pragma unroll
    for (int a = 0; a < 3; ++a)
      res[a] = ed[a] - (R[a][0] * er[0] + R[a][1] * er[1] + R[a][2] * er[2]);
    pv = fmaf(w, res[0] * res[0] + res[1] * res[1] + res[2] * res[2], pv);
    float s = gscale * w;
#pragma unroll
    for (int c = 0; c < 3; ++c) {
      float g = s * res[c];
      atomicAdd(&grad_v[3 * j + c], g);
      acc[c] -= g;
    }
  }
#pragma unroll
  for (int c = 0; c < 3; ++c) atomicAdd(&grad_v[3 * i + c], acc[c]);
  atomicAdd(energy, pv * (1.0f / (float)NVERT));
}

// ---------------------------------------------------------------------------
// backward through layer 2: gh1[i] = relu'(h1) * sum_j W2[i][j] * relu'(h2)*gh2[j]
// ---------------------------------------------------------------------------
__global__ void bwd_mlp2(const float* __restrict__ W2, const float* __restrict__ gh2,
                         const float* __restrict__ h2, const float* __restrict__ h1,
                         float* __restrict__ gh1) {
  int i = blockIdx.x * blockDim.x + threadIdx.x;
  if (i >= H1DIM) return;
  float acc = 0.0f;
  const float* row = W2 + (size_t)i * H2DIM;
  for (int j = 0; j < H2DIM; ++j) {
    float g = (h2[j] > 0.0f) ? gh2[j] : 0.0f;
    acc = fmaf(row[j], g, acc);
  }
  gh1[i] = (h1[i] > 0.0f) ? acc : 0.0f;
}

// ---------------------------------------------------------------------------
// backward through layer 1 + finalize output
// ---------------------------------------------------------------------------
__global__ void bwd_mlp1(const float* __restrict__ W1, const float* __restrict__ gh1,
                         const float* __restrict__ energy, float* __restrict__ out) {
  int c = blockIdx.x * blockDim.x + threadIdx.x;
  if (c < DCODE) {
    float acc = 0.0f;
    const float* row = W1 + (size_t)c * H1DIM;
    for (int i = 0; i < H1DIM; ++i) acc = fmaf(row[i], gh1[i], acc);
    out[1 + c] = acc;                 // dE/dcode
  }
  if (c == 0) out[0] = -energy[0];    // reference returns (-E, grad)
}

// ---------------------------------------------------------------------------
// launch
// ---------------------------------------------------------------------------
extern "C" void kernel_launch(void* const* d_in, const int* in_sizes, int n_in,
                              void* d_out, int out_size, void* d_ws, size_t ws_size,
                              hipStream_t stream) {
  const float* code = (const float*)d_in[0];
  const float* xyz1 = (const float*)d_in[1];
  const float* wmat = (const float*)d_in[2];
  const float* W1   = (const float*)d_in[3];
  const float* b1   = (const float*)d_in[4];
  const float* W2   = (const float*)d_in[5];
  const float* b2   = (const float*)d_in[6];
  const float* W3   = (const float*)d_in[7];
  const float* b3   = (const float*)d_in[8];
  const int*   nbr  = (const int*)d_in[9];
  const int*   nnbr = (const int*)d_in[10];
  float* out = (float*)d_out;
  float* ws  = (float*)d_ws;

  float* h1     = ws + OFF_H1;
  float* h2     = ws + OFF_H2;
  float* recon  = ws + OFF_RECON;
  float* grad_v = ws + OFF_GV;
  float* gh2    = ws + OFF_GH2;
  float* energy = ws + OFF_EN;
  float* gh1    = ws + OFF_GH1;

  // zero grad_v + gh2 + energy (contiguous region)
  const int nzero = M3 + H2DIM + 1;
  zero_kernel<<<(nzero + 255) / 256, 256, 0, stream>>>(grad_v, nzero);

  // forward MLP (small layers)
  mlp_mv_relu<<<(H1DIM + 255) / 256, 256, 0, stream>>>(code, W1, b1, h1, DCODE, H1DIM);
  mlp_mv_relu<<<(H2DIM + 255) / 256, 256, 0, stream>>>(h1, W2, b2, h2, H1DIM, H2DIM);

  // forward big layer via WMMA: 3750 tiles, 8 waves/block
  decode_wmma<<<((M3 / 16) + 7) / 8, 256, 0, stream>>>(h2, W3, b3, recon);

  // ARAP energy + dE/drecon
  arap_kernel<<<(NVERT + 255) / 256, 256, 0, stream>>>(xyz1, recon, wmat, nbr, nnbr,
                                                       grad_v, energy);

  // backward big layer via WMMA: 64 i-tiles x 30 chunks = 1920 waves
  bwd_w3_wmma<<<((H2DIM / 16) * BW_CHUNKS) / 8, 256, 0, stream>>>(W3, grad_v, gh2);

  // backward small layers + output
  bwd_mlp2<<<(H1DIM + 255) / 256, 256, 0, stream>>>(W2, gh2, h2, h1, gh1);
  bwd_mlp1<<<(DCODE + 255) / 256, 256, 0, stream>>>(W1, gh1, energy, out);
}